// DistanceTransform_20340965114029
// MI455X (gfx1250) — compile-verified
//
#include <hip/hip_runtime.h>
#include <stdint.h>

#define H_DIM 256
#define W_DIM 256
#define NPIX (H_DIM * W_DIM)
#define NITER 256            // ceil(max(H,W)/k_half), k_half=1
#define HPAR 0.35f
#define CHUNK_ROWS 16
#define NCHUNK (H_DIM / CHUNK_ROWS)
#define WPR (W_DIM / 32)            // words per row = 8
#define WORDS_TOTAL (H_DIM * WPR)   // 2048
#define WPC (CHUNK_ROWS * WPR)      // words per chunk = 128

#if defined(__gfx1250__) && __has_builtin(__builtin_amdgcn_tensor_load_to_lds) && __has_builtin(__builtin_amdgcn_s_wait_tensorcnt)
#define USE_TDM 1
#else
#define USE_TDM 0
#endif

typedef unsigned int u32x4 __attribute__((ext_vector_type(4)));
typedef int          i32x8 __attribute__((ext_vector_type(8)));
typedef int          i32x4 __attribute__((ext_vector_type(4)));

__device__ __forceinline__ uint32_t ballot32(bool p) {
#if defined(__gfx1250__) && __has_builtin(__builtin_amdgcn_ballot_w32)
  return __builtin_amdgcn_ballot_w32(p);
#else
  return (uint32_t)__ballot(p ? 1 : 0);
#endif
}

#if USE_TDM
// Issue a TDM DMA: 256x16 fp32 tile (one row-chunk of the image) -> LDS stage.
__device__ __forceinline__ void tdm_issue_chunk(const float* src, void* lds_dst, int chunk) {
  uint64_t ga  = (uint64_t)(uintptr_t)(src + (size_t)chunk * CHUNK_ROWS * W_DIM);
  uint32_t lds = (uint32_t)(uintptr_t)lds_dst;  // low 32 bits of flat addr == LDS byte address
  u32x4 g0;
  g0[0] = 1u;                                          // count=1, user descriptor
  g0[1] = lds;                                         // lds_addr
  g0[2] = (uint32_t)ga;                                // global_addr[31:0]
  g0[3] = (uint32_t)((ga >> 32) & 0x01FFFFFFu) | (2u << 30); // global_addr[56:32] | type=2
  i32x8 g1;
  g1[0] = (int)0x00020000u;            // data_size=2 (4 bytes); no mask/pad/iterate
  g1[1] = (int)((uint32_t)W_DIM << 16);      // tensor_dim0 = 256 (bits 79:48 lo)
  g1[2] = (int)((uint32_t)CHUNK_ROWS << 16); // tensor_dim1 = 16
  g1[3] = (int)((uint32_t)W_DIM << 16);      // tile_dim0 = 256
  g1[4] = CHUNK_ROWS;                        // tile_dim1 = 16, tile_dim2 = 0
  g1[5] = W_DIM;                             // tensor_dim0_stride = 256
  g1[6] = 0; g1[7] = 0;
  i32x4 z4 = {0, 0, 0, 0};
  i32x8 z8 = {0, 0, 0, 0, 0, 0, 0, 0};
  // 6-arg form (this toolchain): (g0, g1, g2, g3, g4, cpol)
  __builtin_amdgcn_tensor_load_to_lds(g0, g1, z4, z4, z8, 0);
}
#endif

__global__ __launch_bounds__(1024, 1)
void DistanceTransform_kernel(const float* __restrict__ img, float* __restrict__ out) {
  const int imgIdx = blockIdx.x;
  const float* __restrict__ src = img + (size_t)imgIdx * NPIX;
  float* __restrict__ dst = out + (size_t)imgIdx * NPIX;

  __shared__ uint32_t bnd[WORDS_TOTAL];  // 1 bit per pixel boundary, 8 KB
  __shared__ float lut[32];              // cdt LUT: idx = na*5+nd
  __shared__ int flags[2];               // ping-pong convergence flags
#if USE_TDM
  __shared__ float stage[2][CHUNK_ROWS * W_DIM];  // double-buffered DMA staging, 32 KB
#endif

  const int tid  = threadIdx.x;
  const int lane = tid & 31;
  const int wave = tid >> 5;

  // ---- init: LUT, flags ----
  if (tid < 25) {
    const float A = expf(-1.0f / HPAR);                 // adjacent weight
    const float D = expf(-1.4142135623730951f / HPAR);  // diagonal weight
    int na = tid / 5, nd = tid % 5;
    float conv = (float)na * A + (float)nd * D;
    lut[tid] = (conv > 0.0f) ? (-HPAR * logf(conv)) : 0.0f;
  }
  if (tid == 0) { flags[0] = 0; flags[1] = 0; }

  // ---- zero-init this image's output (each pixel owned by exactly one thread) ----
  {
    float4 z = make_float4(0.0f, 0.0f, 0.0f, 0.0f);
    float4* op = (float4*)(dst + (size_t)tid * 64);
#pragma unroll
    for (int k = 0; k < 16; ++k) op[k] = z;
  }

  // ---- load seed mask into bit-packed boundary ----
#if USE_TDM
  if (wave == 0) tdm_issue_chunk(src, &stage[0][0], 0);
  for (int c = 0; c < NCHUNK; ++c) {
    if (wave == 0) {
      if (c + 1 < NCHUNK) {
        tdm_issue_chunk(src, &stage[(c + 1) & 1][0], c + 1);
        __builtin_amdgcn_s_wait_tensorcnt(1);  // chunk c complete (TDM is in-order)
      } else {
        __builtin_amdgcn_s_wait_tensorcnt(0);
      }
    }
    __syncthreads();  // publish DMA'd LDS data to all waves
    const float* sb = &stage[c & 1][0];
#pragma unroll
    for (int p = 0; p < 4; ++p) {
      int wlocal = wave * 4 + p;  // 0..127 words in this chunk
      float v = sb[wlocal * 32 + lane];
      uint32_t m = ballot32(v > 0.5f);
      if (lane == 0) bnd[c * WPC + wlocal] = m;
    }
    __syncthreads();  // stage buffer may be reused next-next chunk
  }
#else
  for (int p = 0; p < WORDS_TOTAL / 32; ++p) {
    int wi = wave * (WORDS_TOTAL / 32) + p;
    float v = src[(size_t)wi * 32 + lane];
    uint32_t m = ballot32(v > 0.5f);
    if (lane == 0) bnd[wi] = m;
  }
  __syncthreads();
#endif

  // ---- ownership: thread t owns row r = t/4, cols [ (t%4)*64, +64 ) ----
  const int r    = tid >> 2;
  const int q    = tid & 3;
  const int cb   = q << 6;
  const int rowB = r * WPR + (q << 1);
  const int rmB  = ((r == 0)         ? r : r - 1) * WPR + (q << 1);
  const int rpB  = ((r == H_DIM - 1) ? r : r + 1) * WPR + (q << 1);
  const bool hasL = (q != 0);
  const bool hasR = (q != 3);

  // ---- frontier propagation ----
  for (int it = 0; it < NITER; ++it) {
    uint64_t C, Wc, Ec, N, NWb, NEb, S, SWb, SEb;
    {
      uint64_t m  = (uint64_t)bnd[rowB] | ((uint64_t)bnd[rowB + 1] << 32);
      uint64_t lb = hasL ? (uint64_t)(bnd[rowB - 1] >> 31) : (m & 1ull);
      uint64_t rb = hasR ? (uint64_t)(bnd[rowB + 2] & 1u) : ((m >> 63) & 1ull);
      C = m; Wc = (m << 1) | lb; Ec = (m >> 1) | (rb << 63);
    }
    {
      uint64_t m  = (uint64_t)bnd[rmB] | ((uint64_t)bnd[rmB + 1] << 32);
      uint64_t lb = hasL ? (uint64_t)(bnd[rmB - 1] >> 31) : (m & 1ull);
      uint64_t rb = hasR ? (uint64_t)(bnd[rmB + 2] & 1u) : ((m >> 63) & 1ull);
      N = m; NWb = (m << 1) | lb; NEb = (m >> 1) | (rb << 63);
    }
    {
      uint64_t m  = (uint64_t)bnd[rpB] | ((uint64_t)bnd[rpB + 1] << 32);
      uint64_t lb = hasL ? (uint64_t)(bnd[rpB - 1] >> 31) : (m & 1ull);
      uint64_t rb = hasR ? (uint64_t)(bnd[rpB + 2] & 1u) : ((m >> 63) & 1ull);
      S = m; SWb = (m << 1) | lb; SEb = (m >> 1) | (rb << 63);
    }

    uint64_t neigh = N | S | Wc | Ec | NWb | NEb | SWb | SEb;
    uint64_t f = ~C & neigh;  // frontier: center empty, >=1 neighbor set (=> 0 < conv < 1)

    if (f) {
      // SWAR bitplane adders: per-pixel counts of set adjacent / diagonal neighbors
      uint64_t s1 = N ^ S,  c1 = N & S;
      uint64_t s2 = s1 ^ Wc, c2 = s1 & Wc;
      uint64_t p0a = s2 ^ Ec, c3 = s2 & Ec;
      uint64_t t1 = c1 ^ c2, t2 = c1 & c2;
      uint64_t p1a = t1 ^ c3, u2 = t1 & c3;
      uint64_t p2a = t2 | u2;

      uint64_t d1 = NWb ^ NEb, e1 = NWb & NEb;
      uint64_t d2 = d1 ^ SWb,  e2 = d1 & SWb;
      uint64_t p0d = d2 ^ SEb, e3 = d2 & SEb;
      uint64_t g1_ = e1 ^ e2,  g2_ = e1 & e2;
      uint64_t p1d = g1_ ^ e3, h2 = g1_ & e3;
      uint64_t p2d = g2_ | h2;

      const float offset = (float)it;  // i * k_half, k_half = 1
      uint64_t ff = f;
      const int pbase = r * W_DIM + cb;
      while (ff) {
        int j = __ffsll((unsigned long long)ff) - 1;
        ff &= ff - 1;
        int na = (int)((p0a >> j) & 1) | ((int)((p1a >> j) & 1) << 1) | ((int)((p2a >> j) & 1) << 2);
        int nd = (int)((p0d >> j) & 1) | ((int)((p1d >> j) & 1) << 1) | ((int)((p2d >> j) & 1) << 2);
        dst[pbase + j] = offset + lut[na * 5 + nd];  // written exactly once per pixel
      }
    }

    __syncthreads();  // all reads of old boundary complete
    if (f) {
      bnd[rowB]     |= (uint32_t)f;
      bnd[rowB + 1] |= (uint32_t)(f >> 32);
      flags[it & 1] = 1;
    }
    if (tid == 0) flags[(it + 1) & 1] = 0;  // pre-reset next slot (no race: distinct slot)
    __syncthreads();
    if (flags[it & 1] == 0) break;  // uniform: converged (matches reference any(mask))
  }
}

extern "C" void kernel_launch(void* const* d_in, const int* in_sizes, int n_in,
                              void* d_out, int out_size, void* d_ws, size_t ws_size,
                              hipStream_t stream) {
  (void)in_sizes; (void)n_in; (void)d_ws; (void)ws_size; (void)out_size;
  const float* img = (const float*)d_in[0];  // (2,1,256,256) fp32 seed mask
  float* out = (float*)d_out;                // (2,1,256,256) fp32
  DistanceTransform_kernel<<<dim3(2), dim3(1024), 0, stream>>>(img, out);
}